// BaseObservationModel_42923903156755
// MI455X (gfx1250) — compile-verified
//
#include <hip/hip_runtime.h>
#include <stdint.h>

// Problem constants from the reference
#define DIMS   2048          // obs_dim (last axis)
#define KSEL   512           // n masked per row = 0.25 * 2048
#define TPB    256           // threads per block (8 waves of 32 on gfx1250)
#define EPT    8             // elements per thread (2 x 128-bit vectors)
#define NWAVE  (TPB / 32)

// Native clang vector types (HIP_vector_type structs are rejected by the
// nontemporal builtins).
typedef unsigned int u32x4 __attribute__((ext_vector_type(4)));
typedef float        f32x4 __attribute__((ext_vector_type(4)));

// CDNA5 async global->LDS staging (ASYNCcnt) if the toolchain exposes it.
#if defined(__gfx1250__) && \
    __has_builtin(__builtin_amdgcn_global_load_async_to_lds_b128) && \
    __has_builtin(__builtin_amdgcn_s_wait_asynccnt)
#define HAVE_ASYNC_LDS 1
#else
#define HAVE_ASYNC_LDS 0
#endif

#if HAVE_ASYNC_LDS
// Builtin signature (from hipcc diagnostic): pointer params are v4i32* in
// AS1 (global) / AS3 (LDS).
typedef int v4i __attribute__((vector_size(16)));
typedef __attribute__((address_space(1))) v4i* gv4i_p;
typedef __attribute__((address_space(3))) v4i* lv4i_p;
#endif

__global__ __launch_bounds__(TPB) void obs_noise_topk_mask_kernel(
    const float* __restrict__ data,
    const float* __restrict__ noise,
    const float* __restrict__ rand_vals,
    float* __restrict__ out_masked,
    float* __restrict__ out_minv)
{
    const int tid  = threadIdx.x;
    const int lane = tid & 31;
    const int wid  = tid >> 5;
    const long long base = (long long)blockIdx.x * (long long)DIMS;

    __shared__ int s_partial[NWAVE];
    __shared__ int s_total;
    __shared__ int s_scan[TPB];

#if HAVE_ASYNC_LDS
    __shared__ float s_data[DIMS];
    __shared__ float s_noise[DIMS];
    {
        // Kick off async global->LDS B128 copies of this row's data/noise.
        // Each lane stages exactly the 32B it will later consume (no cross-
        // lane LDS sharing, so s_wait_asynccnt 0 alone orders the reads).
        const float* gd = data  + base + tid * EPT;
        const float* gn = noise + base + tid * EPT;
        float* ld = &s_data[tid * EPT];
        float* ln = &s_noise[tid * EPT];
        __builtin_amdgcn_global_load_async_to_lds_b128(
            (gv4i_p)(const void*)gd,       (lv4i_p)(void*)ld,       0, 0);
        __builtin_amdgcn_global_load_async_to_lds_b128(
            (gv4i_p)(const void*)(gd + 4), (lv4i_p)(void*)(ld + 4), 0, 0);
        __builtin_amdgcn_global_load_async_to_lds_b128(
            (gv4i_p)(const void*)gn,       (lv4i_p)(void*)ln,       0, 0);
        __builtin_amdgcn_global_load_async_to_lds_b128(
            (gv4i_p)(const void*)(gn + 4), (lv4i_p)(void*)(ln + 4), 0, 0);
    }
#else
    // Fallback: gfx1250 global_prefetch_b8 so the streaming data is inbound
    // while we run the threshold search.
    __builtin_prefetch(data  + base + tid * EPT, 0, 0);
    __builtin_prefetch(noise + base + tid * EPT, 0, 0);
#endif

    // ---- Load this row's uniform keys (bit pattern order == value order,
    //      since uniforms are non-negative fp32). Two non-temporal B128
    //      loads per thread (single-use stream; keep it out of L2 retention).
    const u32x4* r4 = (const u32x4*)(rand_vals + base);
    u32x4 k0 = __builtin_nontemporal_load(&r4[tid * 2 + 0]);
    u32x4 k1 = __builtin_nontemporal_load(&r4[tid * 2 + 1]);
    uint32_t keys[EPT] = {k0.x, k0.y, k0.z, k0.w, k1.x, k1.y, k1.z, k1.w};

    // ---- Binary search over key bits for T = KSEL-th largest key.
    // Invariant: count(key >= lo) >= KSEL, count(key >= hi) < KSEL.
    // lo = 0 counts all 2048; hi = bits(1.0f) counts 0 (uniforms are < 1).
    uint32_t lo = 0u, hi = 0x3F800000u;
    while (hi - lo > 1u) {
        uint32_t mid = lo + ((hi - lo) >> 1);
        int c = 0;
#pragma unroll
        for (int j = 0; j < EPT; ++j) c += (keys[j] >= mid) ? 1 : 0;
#pragma unroll
        for (int off = 16; off >= 1; off >>= 1) c += __shfl_xor(c, off, 32);
        if (lane == 0) s_partial[wid] = c;
        __syncthreads();
        if (tid == 0) {
            int t = 0;
#pragma unroll
            for (int w = 0; w < NWAVE; ++w) t += s_partial[w];
            s_total = t;
        }
        __syncthreads();
        int cnt = s_total;              // uniform across block
        if (cnt >= KSEL) lo = mid; else hi = mid;
    }
    const uint32_t T = lo;

    // ---- cntGreater = count(key > T); eq = per-thread count(key == T).
    int cg = 0, eq = 0;
#pragma unroll
    for (int j = 0; j < EPT; ++j) {
        cg += (keys[j] > T) ? 1 : 0;
        eq += (keys[j] == T) ? 1 : 0;
    }
    {
        int c = cg;
#pragma unroll
        for (int off = 16; off >= 1; off >>= 1) c += __shfl_xor(c, off, 32);
        if (lane == 0) s_partial[wid] = c;
        __syncthreads();
        if (tid == 0) {
            int t = 0;
#pragma unroll
            for (int w = 0; w < NWAVE; ++w) t += s_partial[w];
            s_total = t;
        }
        __syncthreads();
    }
    const int numTieMask = KSEL - s_total;   // how many ==T elements get masked
                                             // (stable: lowest indices first)

    // ---- Exclusive scan of eq over threads (index order == thread order,
    //      thread t owns contiguous elements [t*8, t*8+8)).
    s_scan[tid] = eq;
    __syncthreads();
    int incl = eq;
    for (int off = 1; off < TPB; off <<= 1) {
        int add = (tid >= off) ? s_scan[tid - off] : 0;
        __syncthreads();
        incl += add;
        s_scan[tid] = incl;
        __syncthreads();
    }
    int tieRank = incl - eq;   // #(key==T) at smaller global indices

    // ---- Finish: noisy = data + 0.1*noise, apply mask, write both outputs.
#if HAVE_ASYNC_LDS
    __builtin_amdgcn_s_wait_asynccnt(0);
    const f32x4* d4 = (const f32x4*)s_data  + tid * 2;
    const f32x4* n4 = (const f32x4*)s_noise + tid * 2;
#else
    const f32x4* d4 = (const f32x4*)(data  + base) + tid * 2;
    const f32x4* n4 = (const f32x4*)(noise + base) + tid * 2;
#endif
    f32x4* om4 = (f32x4*)(out_masked + base) + tid * 2;
    f32x4* oi4 = (f32x4*)(out_minv   + base) + tid * 2;

#pragma unroll
    for (int v = 0; v < 2; ++v) {
        f32x4 dd = d4[v];
        f32x4 nn = n4[v];
        f32x4 om, oi;
#pragma unroll
        for (int j = 0; j < 4; ++j) {
            uint32_t kk = keys[v * 4 + j];
            bool m;
            if (kk > T) {
                m = true;
            } else if (kk == T) {
                m = (tieRank < numTieMask);
                ++tieRank;
            } else {
                m = false;
            }
            float noisy = dd[j] + 0.1f * nn[j];
            om[j] = m ? 0.0f : noisy;
            oi[j] = m ? 0.0f : 1.0f;
        }
        // Streaming outputs: non-temporal stores (single write, never re-read
        // by this dispatch; don't occupy the 192MB L2).
        __builtin_nontemporal_store(om, &om4[v]);
        __builtin_nontemporal_store(oi, &oi4[v]);
    }
}

extern "C" void kernel_launch(void* const* d_in, const int* in_sizes, int n_in,
                              void* d_out, int out_size, void* d_ws, size_t ws_size,
                              hipStream_t stream) {
    (void)n_in; (void)out_size; (void)d_ws; (void)ws_size;
    const float* data      = (const float*)d_in[0];
    const float* noise     = (const float*)d_in[1];
    const float* rand_vals = (const float*)d_in[2];
    float* out = (float*)d_out;

    const long long N    = (long long)in_sizes[0];   // 32*1024*2048
    const long long rows = N / DIMS;                 // 32768

    obs_noise_topk_mask_kernel<<<dim3((unsigned)rows), dim3(TPB), 0, stream>>>(
        data, noise, rand_vals, out /*masked*/, out + N /*mask_inverse*/);
}